// kp_loss_20830591386196
// MI455X (gfx1250) — compile-verified
//
#include <hip/hip_runtime.h>

// MI455X / gfx1250: wave32, WMMA (not MFMA).
// Reference: sum over B*K of exp(-(dx^2+dy^2)/2) / B,  B=8192, K=2048, f32.
// Pure streaming reduction: 256 MiB read @ 23.3 TB/s ~= 11.5 us floor.
// Stage 1: bandwidth-bound grid-stride pass, b128 NT loads, v_exp_f32,
//          wave32 shuffle + LDS tree -> one partial per block (2048 partials).
// Stage 2: single wave reduces 2048 partials deterministically using
//          V_WMMA_F32_16X16X4_F32 with an all-ones B matrix (D = A*1 + C
//          accumulates row sums of 64 partials per WMMA, exact f32 adds).

typedef float __attribute__((ext_vector_type(2))) v2f;
typedef float __attribute__((ext_vector_type(4))) v4f;
typedef float __attribute__((ext_vector_type(8))) v8f;

#define NBLOCKS  2048
#define NTHREADS 256
#define NWAVES   (NTHREADS / 32)

// -0.5 * log2(e): exp(-sq/2) == exp2(sq * C)
#define EXP2_SCALE (-0.72134752044448170368f)

__global__ void __launch_bounds__(NTHREADS)
kp_gauss_partial(const float* __restrict__ outp,
                 const float* __restrict__ tgtp,
                 float* __restrict__ partial,
                 int n4)
{
    const v4f* __restrict__ o4 = (const v4f*)outp;
    const v4f* __restrict__ t4 = (const v4f*)tgtp;

    int tid    = blockIdx.x * NTHREADS + threadIdx.x;
    int stride = gridDim.x * NTHREADS;

    float acc = 0.0f;
    // Each float4 covers 2 keypoints (x,y,x,y). Non-temporal: single-pass
    // stream larger than the 192MB L2, no reuse to cache.
    for (int i = tid; i < n4; i += stride) {
        v4f o = __builtin_nontemporal_load(o4 + i);
        v4f t = __builtin_nontemporal_load(t4 + i);
        float dx0 = o.x - t.x;
        float dy0 = o.y - t.y;
        float dx1 = o.z - t.z;
        float dy1 = o.w - t.w;
        float sq0 = dx0 * dx0 + dy0 * dy0;
        float sq1 = dx1 * dx1 + dy1 * dy1;
        acc += __builtin_amdgcn_exp2f(sq0 * EXP2_SCALE);
        acc += __builtin_amdgcn_exp2f(sq1 * EXP2_SCALE);
    }

    // wave32 butterfly reduction (warpSize == 32 on gfx1250)
    #pragma unroll
    for (int off = 16; off > 0; off >>= 1)
        acc += __shfl_xor(acc, off, 32);

    __shared__ float lds[NWAVES];
    int lane = threadIdx.x & 31;
    int wave = threadIdx.x >> 5;
    if (lane == 0) lds[wave] = acc;
    __syncthreads();

    if (wave == 0) {
        float v = (lane < NWAVES) ? lds[lane] : 0.0f;
        #pragma unroll
        for (int off = NWAVES / 2; off > 0; off >>= 1)
            v += __shfl_xor(v, off, 32);
        if (lane == 0) partial[blockIdx.x] = v;
    }
}

// One wave (32 threads). Reduces `nblocks` (=2048) partials with
// v_wmma_f32_16x16x4_f32: A = 16x4 tile of partials, B = ones(4x16),
// C accumulates row sums replicated across all 16 columns.
// A-layout (ISA 7.12.2): lanes 0-15 hold K=0,1 in VGPR0,1; lanes 16-31 K=2,3.
__global__ void __launch_bounds__(32)
kp_gauss_final(const float* __restrict__ partial,
               float* __restrict__ out,
               float inv_batch,
               int nblocks)
{
    int lane = threadIdx.x;          // full wave, EXEC all-1s for WMMA
    int m    = lane & 15;            // matrix row this lane feeds
    int k0   = (lane >> 4) << 1;     // K pair: lanes 0-15 -> {0,1}, 16-31 -> {2,3}

    v8f c = {};                      // C/D accumulator, starts at 0
    v2f ones;
    ones.x = 1.0f;
    ones.y = 1.0f;                   // all-ones B: layout irrelevant

    for (int base = 0; base < nblocks; base += 64) {
        v2f a;
        a.x = partial[base + m * 4 + k0];
        a.y = partial[base + m * 4 + k0 + 1];
        // D[m][n] = sum_k A[m][k] + C[m][n]   (exact f32, fixed order)
        c = __builtin_amdgcn_wmma_f32_16x16x4_f32(
                /*neg_a=*/false, a, /*neg_b=*/false, ones,
                /*c_mod=*/(short)0, c, /*reuse_a=*/false, /*reuse_b=*/false);
    }

    // Per-lane: sum this lane's 8 C rows. Lane 0 holds M=0..7 (N=0),
    // lane 16 holds M=8..15 (N=0); their sum is the full total.
    float vsum = 0.0f;
    #pragma unroll
    for (int j = 0; j < 8; ++j) vsum += c[j];

    float hi = __shfl(vsum, 16, 32);
    if (lane == 0) out[0] = (vsum + hi) * inv_batch;
}

extern "C" void kernel_launch(void* const* d_in, const int* in_sizes, int n_in,
                              void* d_out, int out_size, void* d_ws, size_t ws_size,
                              hipStream_t stream)
{
    const float* outp = (const float*)d_in[0];   // output: [B,K,2] f32
    const float* tgtp = (const float*)d_in[1];   // target: [B,K,2] f32
    float* partial = (float*)d_ws;               // 2048 floats of scratch
    float* res     = (float*)d_out;

    int n  = in_sizes[0];        // B*K*2 = 33,554,432
    int n4 = n >> 2;             // float4 count (divisible: K*2 % 4 == 0)

    kp_gauss_partial<<<NBLOCKS, NTHREADS, 0, stream>>>(outp, tgtp, partial, n4);

    const float inv_batch = 1.0f / 8192.0f;      // reference batch size B
    kp_gauss_final<<<1, 32, 0, stream>>>(partial, res, inv_batch, NBLOCKS);
}